// MomentumCalcHead_54958401519770
// MI455X (gfx1250) — compile-verified
//
#include <hip/hip_runtime.h>
#include <stdint.h>

// Reference: out[8192,512] = class_sums + segment_sum(batch_samples[131072,512], targets)
// Memory-bound scatter-add. Strategy:
//   K1: out = class_sums (vectorized copy; makes every graph replay idempotent)
//   K2: stream batch rows via CDNA5 async global->LDS copies (double buffered,
//       ASYNCcnt-ordered, no barriers since each lane consumes its own copies),
//       then device-scope f32 atomic adds into out (RMW resolves in 192MB L2;
//       the 16MB output stays L2-resident because the 256MB input stream is
//       marked non-temporal).

#define FEAT_DIM 512

constexpr int ROWS_PER_CHUNK = 8;                               // 8 rows * 2KB = 16KB/chunk
constexpr int CHUNK_FLOATS   = ROWS_PER_CHUNK * FEAT_DIM;       // 4096 floats
constexpr int THREADS        = 256;                             // 8 waves (wave32)
constexpr int V4_PER_THREAD  = (CHUNK_FLOATS / 4) / THREADS;    // 4 float4 per thread

// Per-lane async copy of 16B global -> LDS. Tracked by ASYNCcnt (per-wave,
// in-order completion for loads). NT hint: stream read once, keep L2 for 'out'.
__device__ __forceinline__ void async_f4_to_lds(uint32_t lds_off, const float* gptr) {
  asm volatile("global_load_async_to_lds_b128 %0, %1, off th:TH_LOAD_NT"
               :: "v"(lds_off), "v"((uint64_t)(uintptr_t)gptr)
               : "memory");
}

// Native no-return FP32 atomic add at device scope (meets other WGPs in L2).
__device__ __forceinline__ void global_fadd(float* p, float v) {
  asm volatile("global_atomic_add_f32 %0, %1, off scope:SCOPE_DEV"
               :: "v"((uint64_t)(uintptr_t)p), "v"(v)
               : "memory");
}

__global__ __launch_bounds__(THREADS)
void copy_init_kernel(const float4* __restrict__ src, float4* __restrict__ dst, int n4) {
  int i = blockIdx.x * THREADS + threadIdx.x;
  const int stride = gridDim.x * THREADS;
  for (; i < n4; i += stride) dst[i] = src[i];
}

__global__ __launch_bounds__(THREADS)
void scatter_add_kernel(const float* __restrict__ x,
                        const int* __restrict__ targets,
                        float* __restrict__ out,
                        int num_chunks) {
  __shared__ __align__(16) float buf[2][CHUNK_FLOATS];  // 32KB double buffer
  const int t = threadIdx.x;
  const int gstride = gridDim.x;

  // Each wave issues exactly V4_PER_THREAD (=4) async instructions per chunk.
  auto issue = [&](int chunk, int b) {
    const float* gbase = x + (size_t)chunk * CHUNK_FLOATS;
#pragma unroll
    for (int j = 0; j < V4_PER_THREAD; ++j) {
      const int e = t + j * THREADS;  // float4 slot in chunk; lane copies its own data
      // Flat LDS address: low 32 bits are the LDS byte offset.
      const uint32_t lo = (uint32_t)(uintptr_t)&buf[b][e * 4];
      async_f4_to_lds(lo, gbase + (size_t)e * 4);
    }
  };

  int c = blockIdx.x;
  if (c < num_chunks) issue(c, 0);
  int pb = 0;

  for (; c < num_chunks; c += gstride) {
    const int nc = c + gstride;
    if (nc < num_chunks) {
      issue(nc, pb ^ 1);                               // prefetch next chunk (+4)
      asm volatile("s_wait_asynccnt 4" ::: "memory");  // oldest 4 (= chunk c) done
    } else {
      asm volatile("s_wait_asynccnt 0" ::: "memory");  // drain: chunk c done
    }

    const int row0 = c * ROWS_PER_CHUNK;
#pragma unroll
    for (int j = 0; j < V4_PER_THREAD; ++j) {
      const int e   = t + j * THREADS;
      const int row = e >> 7;              // FEAT_DIM/4 = 128 float4 per row
      const int col = (e & 127) * 4;
      const int cls = targets[row0 + row]; // L2-hot, broadcast-ish
      float* op = out + (size_t)cls * FEAT_DIM + col;
      // Each thread reads back exactly the LDS bytes its own lane copied:
      // per-wave ASYNCcnt wait is sufficient, no workgroup barrier needed.
      const float v0 = buf[pb][e * 4 + 0];
      const float v1 = buf[pb][e * 4 + 1];
      const float v2 = buf[pb][e * 4 + 2];
      const float v3 = buf[pb][e * 4 + 3];
      global_fadd(op + 0, v0);
      global_fadd(op + 1, v1);
      global_fadd(op + 2, v2);
      global_fadd(op + 3, v3);
    }
    pb ^= 1;
  }
}

extern "C" void kernel_launch(void* const* d_in, const int* in_sizes, int n_in,
                              void* d_out, int out_size, void* d_ws, size_t ws_size,
                              hipStream_t stream) {
  const float* batch_samples = (const float*)d_in[0];   // [BATCH, 512] f32
  const float* class_sums    = (const float*)d_in[1];   // [8192, 512] f32
  const int*   targets       = (const int*)d_in[2];     // [BATCH] (int per harness)
  (void)in_sizes; (void)n_in; (void)d_ws; (void)ws_size;
  float* out = (float*)d_out;                           // [8192, 512] f32

  // K1: out = class_sums  (16MB copy, ~1.4us at HBM roofline)
  const int n4 = out_size / 4;
  copy_init_kernel<<<2048, THREADS, 0, stream>>>(
      (const float4*)class_sums, (float4*)out, n4);

  // K2: streaming scatter-add (256MB read -> ~11us at 23.3 TB/s)
  const int batch      = in_sizes[0] / FEAT_DIM;        // 131072
  const int num_chunks = batch / ROWS_PER_CHUNK;        // 16384 (exact)
  int grid = num_chunks < 2048 ? num_chunks : 2048;     // ~8 chunks/block, 2-deep pipe
  scatter_add_kernel<<<grid, THREADS, 0, stream>>>(
      batch_samples, targets, out, num_chunks);
}